// EllipticalShapeLoss_75161927680693
// MI455X (gfx1250) — compile-verified
//
#include <hip/hip_runtime.h>
#include <hip/hip_bf16.h>

typedef __attribute__((ext_vector_type(2))) float v2f;
typedef __attribute__((ext_vector_type(8))) float v8f;

#ifndef FG_H
#define FG_H 1024
#define FG_W 1024
#define FG_B 32
#endif

// ws layout: per batch, 32 doubles; moment M_{p,q} at index p*5+q (p+q<=4 used)

__global__ void esl_zero_ws(double* __restrict__ ws, int n) {
    int i = blockIdx.x * blockDim.x + threadIdx.x;
    if (i < n) ws[i] = 0.0;
}

// Branchless u^m for m in 0..4, zero for m>4 (v_cndmask chain, no EXEC divergence)
__device__ __forceinline__ float esl_pow04(float u, int m) {
    const float u2 = u * u;
    float r = 1.0f;
    r = (m == 1) ? u        : r;
    r = (m == 2) ? u2       : r;
    r = (m == 3) ? u2 * u   : r;
    r = (m == 4) ? u2 * u2  : r;
    r = (m > 4)  ? 0.0f     : r;
    return r;
}

// One workgroup per (64-row tile, batch). 256 threads = 8 waves.
// Phase 1: each wave computes per-row x-moments R[y][0..4] (f32, coords shifted by -512).
// Phase 2: wave 0 does M_{p,q} += sum_y u^p * R[y][q] via V_WMMA_F32_16X16X4_F32,
//          then merges into f64 global accumulators with global_atomic_add_f64.
__global__ __launch_bounds__(256) void esl_moments_kernel(const float* __restrict__ probs,
                                                          double* __restrict__ ws) {
    const int b = blockIdx.y;
    const int rowBase = blockIdx.x * 64;
    const int tid  = threadIdx.x;
    const int lane = tid & 31;
    const int wv   = tid >> 5;

    __shared__ float R[64 * 6];  // 64 rows x (5 moments + zero pad col 5)

    const float* p0b = probs + (size_t)(2 * b + 0) * FG_H * FG_W;
    const float* p1b = probs + (size_t)(2 * b + 1) * FG_H * FG_W;

    // ---- Phase 1: per-row x-moment vectors ----
    for (int i = 0; i < 8; ++i) {
        const int ylocal = wv + 8 * i;
        const int y = rowBase + ylocal;
        const float* r0p = p0b + (size_t)y * FG_W;
        const float* r1p = p1b + (size_t)y * FG_W;

        if (i < 7) {  // prefetch this wave's next row (gfx1250 global_prefetch_b8)
            __builtin_prefetch(r0p + (size_t)8 * FG_W + lane * 32, 0, 0);
            __builtin_prefetch(r1p + (size_t)8 * FG_W + lane * 32, 0, 0);
        }

        float s0 = 0.f, s1 = 0.f, s2 = 0.f, s3 = 0.f, s4 = 0.f;
#pragma unroll
        for (int j = 0; j < 8; ++j) {
            const int x0 = (lane + 32 * j) * 4;            // coalesced b128 loads
            float4 a0 = *reinterpret_cast<const float4*>(r0p + x0);
            float4 a1 = *reinterpret_cast<const float4*>(r1p + x0);
            float c0[4] = {a0.x, a0.y, a0.z, a0.w};
            float c1[4] = {a1.x, a1.y, a1.z, a1.w};
#pragma unroll
            for (int k = 0; k < 4; ++k) {
                const float wgt = (c1[k] > c0[k]) ? 1.0f : 0.0f;  // argmax==FG
                const float xv  = (float)(x0 + k - 512);          // shifted x
                const float x2  = xv * xv;
                s0 += wgt;
                s1 += wgt * xv;
                s2 += wgt * x2;
                s3 += wgt * x2 * xv;
                s4 += wgt * x2 * x2;
            }
        }
        // wave32 reduction
        for (int off = 16; off >= 1; off >>= 1) {
            s0 += __shfl_down(s0, off);
            s1 += __shfl_down(s1, off);
            s2 += __shfl_down(s2, off);
            s3 += __shfl_down(s3, off);
            s4 += __shfl_down(s4, off);
        }
        if (lane == 0) {
            float* Rr = &R[ylocal * 6];
            Rr[0] = s0; Rr[1] = s1; Rr[2] = s2; Rr[3] = s3; Rr[4] = s4; Rr[5] = 0.f;
        }
    }
    __syncthreads();

    // ---- Phase 2: (5x64)*(64x5) via 16 chained f32 WMMAs on wave 0 ----
    if (tid < 32) {
        const int mn   = lane & 15;             // A: M row (power p); B: N col (moment q)
        const int blk  = lane >> 4;             // K block: lanes 16-31 carry K=2,3
        const int mn_c = (mn < 5) ? mn : 5;     // clamp to zero-pad column for n>4
        v8f acc = {};                           // C/D: 16x16 f32 accumulator

        for (int k0 = 0; k0 < 64; k0 += 4) {
            const int kA = k0 + 2 * blk;                      // K for VGPR0; VGPR1 is kA+1
            const float u0 = (float)(rowBase + kA) - 512.0f;  // shifted y
            const float u1 = u0 + 1.0f;
            v2f a, bb;
            a.x  = esl_pow04(u0, mn);                         // rows 5..15 -> 0
            a.y  = esl_pow04(u1, mn);
            bb.x = R[kA * 6 + mn_c];                          // cols 5..15 -> pad 0
            bb.y = R[(kA + 1) * 6 + mn_c];
            // D = A(16x4) * B(4x16) + C   (EXEC all-ones on this wave)
            acc = __builtin_amdgcn_wmma_f32_16x16x4_f32(
                false, a, false, bb, (short)0, acc, false, false);
        }

        // D layout: lanes 0-15 -> N=lane, VGPR j -> M=j. We need p,q in 0..4, p+q<=4.
        if (lane < 5) {
#pragma unroll
            for (int p = 0; p <= 4; ++p) {
                if (p + lane <= 4) {
                    atomicAdd(&ws[b * 32 + p * 5 + lane], (double)acc[p]);
                }
            }
        }
    }
}

// One wave: one batch per lane, then wave-reduce, lane0 writes scalar.
__global__ void esl_finalize_kernel(const double* __restrict__ ws, float* __restrict__ out) {
    const int b = threadIdx.x;  // 0..31
    const double* M = ws + b * 32;
    const double M00 = M[0],  M01 = M[1],  M02 = M[2],  M03 = M[3],  M04 = M[4];
    const double M10 = M[5],  M11 = M[6],  M12 = M[7],  M13 = M[8];
    const double M20 = M[10], M21 = M[11], M22 = M[12];
    const double M30 = M[15], M31 = M[16];
    const double M40 = M[20];

    const double n      = M00;
    const double n_safe = fmax(n, 2.0);
    const double denomc = fmax(n_safe - 1.0, 1.0);
    // centroid offsets in the shifted frame (handles n_safe != n edge case)
    const double al = M10 / n_safe + 512.0 * (M00 / n_safe - 1.0);
    const double be = M01 / n_safe + 512.0 * (M00 / n_safe - 1.0);
    const double al2 = al * al, al3 = al2 * al, al4 = al3 * al;
    const double be2 = be * be, be3 = be2 * be, be4 = be3 * be;

    // central moment sums S_{p,q} = sum w (u-al)^p (v-be)^q
    const double S20 = M20 - 2.0 * al * M10 + al2 * M00;
    const double S11 = M11 - al * M01 - be * M10 + al * be * M00;
    const double S02 = M02 - 2.0 * be * M01 + be2 * M00;
    const double S40 = M40 - 4.0 * al * M30 + 6.0 * al2 * M20 - 4.0 * al3 * M10 + al4 * M00;
    const double S31 = M31 - be * M30 - 3.0 * al * M21 + 3.0 * al * be * M20
                     + 3.0 * al2 * M11 - 3.0 * al2 * be * M10 - al3 * M01 + al3 * be * M00;
    const double S22 = M22 - 2.0 * be * M21 + be2 * M20 - 2.0 * al * M12
                     + 4.0 * al * be * M11 - 2.0 * al * be2 * M10 + al2 * M02
                     - 2.0 * al2 * be * M01 + al2 * be2 * M00;
    const double S13 = M13 - al * M03 - 3.0 * be * M12 + 3.0 * al * be * M02
                     + 3.0 * be2 * M11 - 3.0 * al * be2 * M01 - be3 * M10 + al * be3 * M00;
    const double S04 = M04 - 4.0 * be * M03 + 6.0 * be2 * M02 - 4.0 * be3 * M01 + be4 * M00;

    const double Syy = S20 / denomc, Sxy = S11 / denomc, Sxx = S02 / denomc;
    const double A = Syy + 1e-6, D = Sxx + 1e-6, Bc = Sxy;
    const double det = A * D - Bc * Bc;
    const double i00 = D / det, i11 = A / det, i01 = -Bc / det;

    const double sum_m  = i00 * S20 + 2.0 * i01 * S11 + i11 * S02;
    const double sum_m2 = i00 * i00 * S40 + 4.0 * i00 * i01 * S31
                        + (2.0 * i00 * i11 + 4.0 * i01 * i01) * S22
                        + 4.0 * i01 * i11 * S13 + i11 * i11 * S04;
    const double loss_obj = (sum_m2 - 2.0 * sum_m + n) / n_safe;

    const double valid = (n >= 10.0) ? 1.0 : 0.0;
    double total = valid * loss_obj;
    double cnt   = valid;
    for (int off = 16; off >= 1; off >>= 1) {
        total += __shfl_down(total, off);
        cnt   += __shfl_down(cnt, off);
    }
    if (b == 0) out[0] = (float)((cnt > 0.0) ? (total / fmax(cnt, 1.0)) : 0.0);
}

extern "C" void kernel_launch(void* const* d_in, const int* in_sizes, int n_in,
                              void* d_out, int out_size, void* d_ws, size_t ws_size,
                              hipStream_t stream) {
    const float* probs = (const float*)d_in[0];  // (32, 2, 1024, 1024) f32
    float* out = (float*)d_out;                  // scalar f32
    double* ws = (double*)d_ws;                  // 32 batches * 32 doubles = 8 KB

    esl_zero_ws<<<4, 256, 0, stream>>>(ws, FG_B * 32);

    dim3 grid(FG_H / 64, FG_B);                  // 16 row-tiles x 32 batches
    esl_moments_kernel<<<grid, 256, 0, stream>>>(probs, ws);

    esl_finalize_kernel<<<1, 32, 0, stream>>>(ws, out);
}